// ManyToManyRNN_27547920236740
// MI455X (gfx1250) — compile-verified
//
#include <hip/hip_runtime.h>

typedef float v2f __attribute__((ext_vector_type(2)));
typedef float v4f __attribute__((ext_vector_type(4)));
typedef float v8f __attribute__((ext_vector_type(8)));

static constexpr int Bsz = 4096;
static constexpr int Tsz = 2048;
static constexpr int Hsz = 10;

// Swap the two 16-lane halves of a wave32 (identity lane-select).
__device__ __forceinline__ float swap16(float v) {
#if __has_builtin(__builtin_amdgcn_permlanex16)
  unsigned u = __float_as_uint(v);
  unsigned r = __builtin_amdgcn_permlanex16(u, u, 0x76543210u, 0xfedcba98u,
                                            false, false);
  return __uint_as_float(r);
#else
  return __shfl_xor(v, 16, 32);
#endif
}

__device__ __forceinline__ float fast_tanh(float x) {
#if __has_builtin(__builtin_amdgcn_tanhf)
  return __builtin_amdgcn_tanhf(x);
#else
  float xc = fminf(fmaxf(x, -15.0f), 15.0f);
  float e  = __expf(2.0f * xc);
  return (e - 1.0f) / (e + 1.0f);
#endif
}

__device__ __forceinline__ v8f wmma4(v2f a, v2f b, v8f c) {
  // (neg_a, A, neg_b, B, c_mod, C, reuse_a, reuse_b)
  return __builtin_amdgcn_wmma_f32_16x16x4_f32(false, a, false, b, (short)0, c,
                                               false, false);
}

__global__ void __launch_bounds__(32)
rnn_wmma_kernel(const float* __restrict__ x,     // [B,T]
                const float* __restrict__ w_ih,  // [H,1]
                const float* __restrict__ w_hh,  // [H,H]
                const float* __restrict__ b_ih,  // [H]
                const float* __restrict__ b_hh,  // [H]
                const float* __restrict__ fc_w,  // [1,H]
                const float* __restrict__ fc_b,  // [1]
                float* __restrict__ out) {       // [B,T]
  const int lane = threadIdx.x & 31;
  const int m    = lane & 15;        // row within tile (A: M, C/D: N=batch col)
  const int hi   = (lane >> 4) & 1;  // which 16-lane half
  const int bb   = blockIdx.x * 16;  // batch tile base

  // ---- A operand: padded w_hh, constant over all timesteps ----------------
  // A layout (16x4 f32): lane m holds row m; VGPR j holds K = 4c + j + 2*hi.
  v2f a0, a1, a2;
  {
    int k00 = 0 + 2 * hi, k01 = 1 + 2 * hi;
    int k10 = 4 + 2 * hi, k11 = 5 + 2 * hi;
    int k20 = 8 + 2 * hi, k21 = 9 + 2 * hi;
    a0.x = (m < Hsz && k00 < Hsz) ? w_hh[m * Hsz + k00] : 0.0f;
    a0.y = (m < Hsz && k01 < Hsz) ? w_hh[m * Hsz + k01] : 0.0f;
    a1.x = (m < Hsz && k10 < Hsz) ? w_hh[m * Hsz + k10] : 0.0f;
    a1.y = (m < Hsz && k11 < Hsz) ? w_hh[m * Hsz + k11] : 0.0f;
    a2.x = (m < Hsz && k20 < Hsz) ? w_hh[m * Hsz + k20] : 0.0f;
    a2.y = (m < Hsz && k21 < Hsz) ? w_hh[m * Hsz + k21] : 0.0f;
  }

  // ---- per-D-row constants (row r = g + 8*hi) -----------------------------
  float wrow[8], brow[8], fcrow[8];
#pragma unroll
  for (int g = 0; g < 8; ++g) {
    int r   = g + 8 * hi;
    bool ok = (r < Hsz);
    wrow[g]  = ok ? w_ih[r] : 0.0f;
    brow[g]  = ok ? (b_ih[r] + b_hh[r]) : 0.0f;
    fcrow[g] = ok ? fc_w[r] : 0.0f;
  }
  const float fcb = fc_b[0];

  // State S = h^T (hidden x batch) held in C/D layout: VGPR g = rows g, g+8.
  v8f S = {};

  const bool lo = (hi == 0);
  const float* xp = x   + (size_t)(bb + m) * Tsz;
  float*       op = out + (size_t)(bb + m) * Tsz;

  v4f xcur = *(const v4f*)(xp);
  for (int t = 0; t < Tsz; t += 4) {
    int tn = (t + 4 < Tsz) ? (t + 4) : t;
    v4f xnext = *(const v4f*)(xp + tn);  // prefetch next group
    v4f o;
#pragma unroll
    for (int s = 0; s < 4; ++s) {
      float xv = xcur[s];

      // C = xw_t + biases, built directly in C-layout.
      v8f c;
#pragma unroll
      for (int g = 0; g < 8; ++g) c[g] = fmaf(xv, wrow[g], brow[g]);

      // D-layout state -> B-layout chunks (B 4x16: VGPR0 = k,k+2; VGPR1 = k+1,k+3
      // with low lanes = low-k pair, high lanes = high-k pair).
      float X0 = swap16(S[0]);
      float X1 = swap16(S[1]);
      float X2 = swap16(S[2]);
      float X3 = swap16(S[3]);
      float X6 = swap16(S[6]);
      float X7 = swap16(S[7]);
      v2f b0, b1, b2;
      b0.x = lo ? S[0] : X2;   // k=0 | k=2
      b0.y = lo ? S[1] : X3;   // k=1 | k=3
      b1.x = lo ? S[4] : X6;   // k=4 | k=6
      b1.y = lo ? S[5] : X7;   // k=5 | k=7
      b2.x = lo ? X0 : S[2];   // k=8 | k=10
      b2.y = lo ? X1 : S[3];   // k=9 | k=11
      // k=12..15 chunk is all-zero (H=10) -> skipped.

      v8f d = wmma4(a0, b0, c);
      d = wmma4(a1, b1, d);
      d = wmma4(a2, b2, d);

#pragma unroll
      for (int g = 0; g < 8; ++g) S[g] = fast_tanh(d[g]);

      // FC: per-half partial dot, then cross-half reduce (all lanes get sum).
      float acc = 0.0f;
#pragma unroll
      for (int g = 0; g < 8; ++g) acc = fmaf(S[g], fcrow[g], acc);
      acc = acc + swap16(acc);
      o[s] = acc + fcb;
    }
    // Lanes n and n+16 hold identical values & addresses -> benign dup store,
    // keeps EXEC all-ones around the WMMA ops.
    *(v4f*)(op + t) = o;
    xcur = xnext;
  }
}

extern "C" void kernel_launch(void* const* d_in, const int* in_sizes, int n_in,
                              void* d_out, int out_size, void* d_ws, size_t ws_size,
                              hipStream_t stream) {
  (void)in_sizes; (void)n_in; (void)out_size; (void)d_ws; (void)ws_size;
  const float* x    = (const float*)d_in[0];
  const float* w_ih = (const float*)d_in[1];
  const float* w_hh = (const float*)d_in[2];
  const float* b_ih = (const float*)d_in[3];
  const float* b_hh = (const float*)d_in[4];
  const float* fc_w = (const float*)d_in[5];
  const float* fc_b = (const float*)d_in[6];
  float* out = (float*)d_out;

  dim3 grid(Bsz / 16);  // 256 tiles, one wave32 each
  dim3 block(32);
  rnn_wmma_kernel<<<grid, block, 0, stream>>>(x, w_ih, w_hh, b_ih, b_hh,
                                              fc_w, fc_b, out);
}